// CenterLoss_back_89326729822504
// MI455X (gfx1250) — compile-verified
//
#include <hip/hip_runtime.h>
#include <hip/hip_bf16.h>

// CenterLoss forward: mean over batch of clipped squared distance to label's center.
//  x:       [256, 512] f32
//  labels:  [256]      i64
//  centers: [85742,512] f32
//  out:     [1] f32
//
// Strategy: per 16-row tile, dist = diag( d * d^T ) with d = x - centers[labels],
// computed on the CDNA5 matrix pipe via V_WMMA_F32_16X16X4_F32 accumulating over
// D=512 in K=4 slices. For this instruction the A-fragment and the B-fragment of
// A's transpose have identical per-lane register layouts, so the same v2f is fed
// to both operands. f32 multiply / f32 accumulate => reference precision.

typedef __attribute__((ext_vector_type(2))) float v2f;
typedef __attribute__((ext_vector_type(8))) float v8f;

#define FEAT_DIM 512
#define BATCH    256
#define THREADS  512   // 16 wave32s, one 16-row tile per wave

__global__ __launch_bounds__(THREADS) void center_loss_mean_kernel(
    const float* __restrict__ x,
    const long long* __restrict__ labels,
    const float* __restrict__ centers,
    float* __restrict__ out) {

  __shared__ float s_partial[16];

  const int tid  = threadIdx.x;
  const int lane = tid & 31;
  const int wave = tid >> 5;       // tile index 0..15
  const int m    = lane & 15;      // row within tile (A-matrix M = lane%16)
  const int kh   = lane >> 4;      // K half: lanes 0-15 -> K0,K1 ; lanes 16-31 -> K2,K3

  const int row = wave * 16 + m;
  const long long lab = labels[row];
  const float* __restrict__ xr = x + (size_t)row * FEAT_DIM;
  const float* __restrict__ cr = centers + (size_t)lab * FEAT_DIM;

  v8f acc = {};   // 16x16 f32 Gram accumulator

  // 128 K=4 slices over the feature dimension. Each lane fetches the two
  // consecutive columns its fragment slot requires; across the two lane
  // halves every column is read exactly once.
#pragma unroll 8
  for (int k0 = 0; k0 < FEAT_DIM; k0 += 4) {
    const int col = k0 + 2 * kh;
    const float2 xv = *(const float2*)(xr + col);
    const float2 cv = *(const float2*)(cr + col);
    v2f a;
    a.x = xv.x - cv.x;
    a.y = xv.y - cv.y;
    // D = A * A^T + C : same fragment serves as A and as B (=A^T).
    // args: (neg_a, A, neg_b, B, c_mod, C, reuse_a, reuse_b)
    acc = __builtin_amdgcn_wmma_f32_16x16x4_f32(
        false, a, false, a, (short)0, acc, false, false);
  }

  // Diagonal extraction from the 16x16 f32 C/D layout:
  //   VGPR g : lanes 0-15 -> C[g][lane], lanes 16-31 -> C[8+g][lane-16]
  // => lane g (g<8) holds C[g][g] in acc[g]; lane 24+g holds C[8+g][8+g] in acc[g].
  const int idx = (lane < 16) ? lane : (lane - 24);
  float dv = 0.0f;
#pragma unroll
  for (int g = 0; g < 8; ++g) {
    dv = (idx == g) ? acc[g] : dv;   // lanes 8..23 never match -> contribute 0
  }
  const bool valid = (lane < 8) || (lane >= 24);
  if (valid) {
    dv = fminf(fmaxf(dv, 1e-12f), 1e12f);  // clip(1e-12, 1e12)
  }

  // Wave32 reduction of the 16 row-distances held in this wave.
#pragma unroll
  for (int off = 16; off > 0; off >>= 1) {
    dv += __shfl_xor(dv, off, 32);
  }
  if (lane == 0) s_partial[wave] = dv;
  __syncthreads();

  if (tid == 0) {
    float total = 0.0f;
#pragma unroll
    for (int w = 0; w < 16; ++w) total += s_partial[w];
    out[0] = total * (1.0f / (float)BATCH);
  }
}

extern "C" void kernel_launch(void* const* d_in, const int* in_sizes, int n_in,
                              void* d_out, int out_size, void* d_ws, size_t ws_size,
                              hipStream_t stream) {
  (void)in_sizes; (void)n_in; (void)out_size; (void)d_ws; (void)ws_size;
  const float*     x       = (const float*)d_in[0];
  const long long* labels  = (const long long*)d_in[1];
  const float*     centers = (const float*)d_in[2];
  float*           out     = (float*)d_out;

  center_loss_mean_kernel<<<dim3(1), dim3(THREADS), 0, stream>>>(x, labels, centers, out);
}